// PraxisScatter_20710332302090
// MI455X (gfx1250) — compile-verified
//
#include <hip/hip_runtime.h>
#include <stdint.h>

// ---------------------------------------------------------------------------
// Problem constants (match reference)
// ---------------------------------------------------------------------------
constexpr int Bb = 16;          // batch
constexpr int Ss = 512;         // seq
constexpr int Dd = 1024;        // model dim
constexpr int Hh = 4096;        // hidden dim
constexpr int BS = Bb * Ss;     // 8192 flattened rows
constexpr unsigned KSEL = (Hh / 4) * Ss;  // 524288 = top-k per batch

// GEMM tiling: 128x128 block, 8 wave32 waves, each wave 32x64 (2x4 WMMA tiles)
constexpr int BM = 128;
constexpr int BN = 128;
constexpr int BK = 16;
constexpr int LDT = BK + 4;     // padded LDS stride (multiple of 4 -> 16B-aligned float4)

typedef __attribute__((ext_vector_type(2))) float v2f;
typedef __attribute__((ext_vector_type(8))) float v8f;

// Monotone float -> uint32 key (float order == unsigned order)
__device__ __forceinline__ unsigned fkey(float f) {
    unsigned u = __float_as_uint(f);
    return u ^ ((u >> 31) ? 0xFFFFFFFFu : 0x80000000u);
}

// ---------------------------------------------------------------------------
// CDNA5 async global->LDS copy path (ASYNCcnt), with fallback
// ---------------------------------------------------------------------------
#if defined(__AMDGCN__) && __has_builtin(__builtin_amdgcn_global_load_async_to_lds_b128)
#define USE_ASYNC 1
#else
#define USE_ASYNC 0
#endif

#if USE_ASYNC
// Builtin signature (from hipcc diagnostic): param 0 is a non-const pointer to
// a 16-byte int vector in the global address space; param 1 is the LDS side.
typedef int v4i_g __attribute__((vector_size(16)));
typedef __attribute__((address_space(1))) v4i_g GV4;
typedef __attribute__((address_space(3))) v4i_g LV4;
__device__ __forceinline__ void cp16_async(float* lds, const float* g) {
    __builtin_amdgcn_global_load_async_to_lds_b128((GV4*)g, (LV4*)lds, 0, 0);
}
#if __has_builtin(__builtin_amdgcn_s_wait_asynccnt)
#define WAIT_ASYNC(n) __builtin_amdgcn_s_wait_asynccnt(n)
#else
#define WAIT_ASYNC(n) asm volatile("s_wait_asynccnt %0" ::"i"(n) : "memory")
#endif
#endif

// ---------------------------------------------------------------------------
// fp32 WMMA GEMM:  C[M,N] = act(A[M,K] @ Wsel[N,K]^T + bias_sel[N])
// UPMOD=true: per-row weight/bias select from mask (batch-local), ACT=2 = SiLU
// ---------------------------------------------------------------------------
template <int ACT, bool UPMOD>
__global__ __launch_bounds__(256) void k_gemm(
    float* __restrict__ C, const float* __restrict__ A,
    const float* __restrict__ W0, const float* __restrict__ W1,
    const float* __restrict__ bias0, const float* __restrict__ bias1,
    const uint8_t* __restrict__ mask, int M, int N, int K)
{
    __shared__ float As[2][BM * LDT];
    __shared__ float Wsh[2][BN * LDT];

    const int t    = threadIdx.x;
    const int lane = t & 31;
    const int wave = t >> 5;
    const int wm   = wave & 3;      // 0..3 -> 32-row group
    const int wn   = wave >> 2;     // 0..1 -> 64-col group
    const int lid  = lane & 15;
    const int half = lane >> 4;

    const int m0 = blockIdx.y * BM;
    const int n0 = blockIdx.x * BN;
    const uint8_t* mrow = UPMOD ? (mask + (size_t)(m0 >> 9) * Hh) : nullptr;

    // staging coordinates: A tile 128x16 = 512 float4 (2/thread); W tile same
    int fr[2], fc[2];
#pragma unroll
    for (int i = 0; i < 2; ++i) { int f = t + i * 256; fr[i] = f >> 2; fc[i] = (f & 3) << 2; }

    const float* wsrc[2];
#pragma unroll
    for (int i = 0; i < 2; ++i) {
        if constexpr (UPMOD) {
            wsrc[i] = mrow[n0 + fr[i]] ? W1 : W0;   // row-level up/mod select
        } else {
            wsrc[i] = W0;
        }
    }

    v8f acc[2][4] = {};

    auto compute = [&](int buf) {
#pragma unroll
        for (int kk = 0; kk < BK; kk += 4) {
            const int kb = kk + 2 * half;
            v2f a[2], b[4];
#pragma unroll
            for (int i = 0; i < 2; ++i)
                a[i] = *reinterpret_cast<const v2f*>(&As[buf][(32 * wm + 16 * i + lid) * LDT + kb]);
#pragma unroll
            for (int j = 0; j < 4; ++j)
                b[j] = *reinterpret_cast<const v2f*>(&Wsh[buf][(64 * wn + 16 * j + lid) * LDT + kb]);
#pragma unroll
            for (int i = 0; i < 2; ++i)
#pragma unroll
                for (int j = 0; j < 4; ++j)
                    acc[i][j] = __builtin_amdgcn_wmma_f32_16x16x4_f32(
                        false, a[i], false, b[j], (short)0, acc[i][j], false, false);
        }
    };

#if USE_ASYNC
    auto stage = [&](int buf, int k0) {
#pragma unroll
        for (int i = 0; i < 2; ++i)
            cp16_async(&As[buf][fr[i] * LDT + fc[i]],
                       &A[(size_t)(m0 + fr[i]) * K + k0 + fc[i]]);
#pragma unroll
        for (int i = 0; i < 2; ++i)
            cp16_async(&Wsh[buf][fr[i] * LDT + fc[i]],
                       &wsrc[i][(size_t)(n0 + fr[i]) * K + k0 + fc[i]]);
    };
    stage(0, 0);
    int cur = 0;
    for (int k0 = 0; k0 < K; k0 += BK) {
        __syncthreads();                        // safe to overwrite buf cur^1
        if (k0 + BK < K) { stage(cur ^ 1, k0 + BK); WAIT_ASYNC(4); }
        else             { WAIT_ASYNC(0); }
        __syncthreads();                        // all waves' tile data landed
        compute(cur);
        cur ^= 1;
    }
#else
    float4 ra[2], rw[2];
    auto gload = [&](int k0) {
#pragma unroll
        for (int i = 0; i < 2; ++i)
            ra[i] = *reinterpret_cast<const float4*>(&A[(size_t)(m0 + fr[i]) * K + k0 + fc[i]]);
#pragma unroll
        for (int i = 0; i < 2; ++i)
            rw[i] = *reinterpret_cast<const float4*>(&wsrc[i][(size_t)(n0 + fr[i]) * K + k0 + fc[i]]);
    };
    auto lstore = [&](int buf) {
#pragma unroll
        for (int i = 0; i < 2; ++i)
            *reinterpret_cast<float4*>(&As[buf][fr[i] * LDT + fc[i]]) = ra[i];
#pragma unroll
        for (int i = 0; i < 2; ++i)
            *reinterpret_cast<float4*>(&Wsh[buf][fr[i] * LDT + fc[i]]) = rw[i];
    };
    gload(0); lstore(0);
    int cur = 0;
    for (int k0 = 0; k0 < K; k0 += BK) {
        if (k0 + BK < K) gload(k0 + BK);        // hide global latency under compute
        __syncthreads();                        // buf[cur] visible; prev reads done
        compute(cur);
        if (k0 + BK < K) lstore(cur ^ 1);
        cur ^= 1;
    }
#endif

    // Epilogue. C/D layout: VGPR r, lane L -> row = r + 8*(L/16), col = L%16
#pragma unroll
    for (int j = 0; j < 4; ++j) {
        const int col = n0 + 64 * wn + 16 * j + lid;
        float bv;
        if constexpr (UPMOD) bv = mrow[col] ? bias1[col] : bias0[col];
        else                 bv = bias0[col];
#pragma unroll
        for (int i = 0; i < 2; ++i)
#pragma unroll
            for (int r = 0; r < 8; ++r) {
                const int row = m0 + 32 * wm + 16 * i + r + 8 * half;
                float v = acc[i][j][r] + bv;
                if (ACT == 1) v = fmaxf(v, 0.0f);
                if (ACT == 2) v = v / (1.0f + __expf(-v));   // SiLU
                C[(size_t)row * N + col] = v;
            }
    }
}

// ---------------------------------------------------------------------------
// Exact per-batch k-th-largest via 4-level radix select on monotone keys
// ---------------------------------------------------------------------------
__global__ void k_init_sel(unsigned* prefix, unsigned* krem) {
    int b = threadIdx.x;
    if (b < Bb) { prefix[b] = 0u; krem[b] = KSEL; }
}

__global__ void k_zero_hist(unsigned* hist) {
    hist[blockIdx.x * 256 + threadIdx.x] = 0u;
}

__global__ __launch_bounds__(256) void k_hist(
    const float* __restrict__ scores, unsigned* __restrict__ hist,
    const unsigned* __restrict__ prefix, int level)
{
    __shared__ unsigned lh[256];
    const int b = blockIdx.y;
    lh[threadIdx.x] = 0u;
    __syncthreads();

    const unsigned pre = prefix[b];
    const int shift = 24 - 8 * level;
    const float* base = scores + (size_t)b * (Ss * Hh);
    const int n = Ss * Hh;
    const int stride = gridDim.x * blockDim.x;
    for (int i = blockIdx.x * blockDim.x + threadIdx.x; i < n; i += stride) {
        unsigned key = fkey(base[i]);
        bool ok = (level == 0) || ((key >> (shift + 8)) == pre);
        if (ok) atomicAdd(&lh[(key >> shift) & 255u], 1u);
    }
    __syncthreads();
    atomicAdd(&hist[b * 256 + threadIdx.x], lh[threadIdx.x]);
}

__global__ void k_scan(const unsigned* __restrict__ hist,
                       unsigned* __restrict__ prefix,
                       unsigned* __restrict__ krem)
{
    if (threadIdx.x != 0) return;
    const int b = blockIdx.x;
    unsigned kr = krem[b];
    unsigned cum = 0;
    int bin = 0;
    for (int i = 255; i >= 0; --i) {
        unsigned c = hist[b * 256 + i];
        if (cum + c >= kr) { bin = i; break; }
        cum += c;
    }
    prefix[b] = (prefix[b] << 8) | (unsigned)bin;
    krem[b] = kr - cum;
}

// mask[b,h] = max_s key(scores[b,s,h]) >= threshold_key[b]
__global__ __launch_bounds__(256) void k_mask(
    const float* __restrict__ scores, const unsigned* __restrict__ thr,
    uint8_t* __restrict__ mask)
{
    const int idx = blockIdx.x * blockDim.x + threadIdx.x;   // 0..B*H
    const int b = idx >> 12;            // H = 4096
    const int h = idx & (Hh - 1);
    const float* base = scores + (size_t)b * (Ss * Hh) + h;
    unsigned mx = 0u;
    for (int s = 0; s < Ss; ++s) {
        unsigned k = fkey(base[(size_t)s * Hh]);
        mx = (k > mx) ? k : mx;
    }
    mask[idx] = (mx >= thr[b]) ? 1u : 0u;
}

// ---------------------------------------------------------------------------
// Driver
// ---------------------------------------------------------------------------
extern "C" void kernel_launch(void* const* d_in, const int* in_sizes, int n_in,
                              void* d_out, int out_size, void* d_ws, size_t ws_size,
                              hipStream_t stream) {
    const float* inputs  = (const float*)d_in[0];
    const float* up_w    = (const float*)d_in[1];
    const float* up_b    = (const float*)d_in[2];
    const float* gate_w1 = (const float*)d_in[3];
    const float* gate_b1 = (const float*)d_in[4];
    const float* gate_w2 = (const float*)d_in[5];
    const float* gate_b2 = (const float*)d_in[6];
    const float* mod_w   = (const float*)d_in[7];
    const float* mod_b   = (const float*)d_in[8];
    const float* down_w  = (const float*)d_in[9];
    const float* down_b  = (const float*)d_in[10];
    float* out = (float*)d_out;

    char* ws = (char*)d_ws;
    const size_t SCORES_BYTES = (size_t)BS * Hh * sizeof(float);   // 134 MB
    float*    scores1 = (float*)ws;
    float*    scores2 = (float*)(ws + SCORES_BYTES);
    float*    hbuf    = scores1;                                   // reuse after gemm2
    unsigned* hist    = (unsigned*)(ws + 2 * SCORES_BYTES);
    unsigned* prefix  = hist + Bb * 256;
    unsigned* krem    = prefix + Bb;
    uint8_t*  mask    = (uint8_t*)(krem + Bb);

    const dim3 blk(256);

    // 1) scores1 = relu(X @ gate_w1^T + gate_b1)      [8192,4096]
    k_gemm<1, false><<<dim3(Hh / BN, BS / BM), blk, 0, stream>>>(
        scores1, inputs, gate_w1, nullptr, gate_b1, nullptr, nullptr, BS, Hh, Dd);

    // 2) scores2 = scores1 @ gate_w2^T + gate_b2      [8192,4096]
    k_gemm<0, false><<<dim3(Hh / BN, BS / BM), blk, 0, stream>>>(
        scores2, scores1, gate_w2, nullptr, gate_b2, nullptr, nullptr, BS, Hh, Hh);

    // 3) per-batch k-th largest (radix select, exact)
    k_init_sel<<<1, Bb, 0, stream>>>(prefix, krem);
    for (int level = 0; level < 4; ++level) {
        k_zero_hist<<<Bb, 256, 0, stream>>>(hist);
        k_hist<<<dim3(128, Bb), blk, 0, stream>>>(scores2, hist, prefix, level);
        k_scan<<<Bb, 32, 0, stream>>>(hist, prefix, krem);
    }

    // 4) mask[b,h]
    k_mask<<<(Bb * Hh) / 256, blk, 0, stream>>>(scores2, prefix, mask);

    // 5) hbuf = silu(X @ select(mask, mod_w, up_w)^T + select-bias)
    k_gemm<2, true><<<dim3(Hh / BN, BS / BM), blk, 0, stream>>>(
        hbuf, inputs, up_w, mod_w, up_b, mod_b, mask, BS, Hh, Dd);

    // 6) out = hbuf @ down_w^T + down_b               [8192,1024]
    k_gemm<0, false><<<dim3(Dd / BN, BS / BM), blk, 0, stream>>>(
        out, hbuf, down_w, nullptr, down_b, nullptr, nullptr, BS, Dd, Hh);
}